// NTXentLoss_23673859736135
// MI455X (gfx1250) — compile-verified
//
#include <hip/hip_runtime.h>

// ---------------------------------------------------------------------------
// Supervised contrastive (NT-Xent) loss, N=8192, D=256, L=16 labels.
// sim = E E^T / T via V_WMMA_F32_16X16X4_F32 tiles, fused with a single-sweep
// ONLINE softmax (flash-style).  B fragments are software-pipelined 8 deep
// with cross-tile prefetch so the load pipe never drains across the J sweep.
// ---------------------------------------------------------------------------

typedef float v2f __attribute__((ext_vector_type(2)));
typedef float v8f __attribute__((ext_vector_type(8)));

static constexpr int D = 256;          // embedding dim (K of the GEMM)
static constexpr int L = 16;           // labels per row
static constexpr int KSTEPS = D / 4;   // 64 wmma_f32_16x16x4 steps per tile
static constexpr int PIPE = 8;         // B-fragment pipeline depth
static constexpr float TEMP_INV = 1.0f / 0.07f;

// ---------------------------------------------------------------------------
// Kernel 0: pack multi-hot int32 labels into 16-bit masks.
// ---------------------------------------------------------------------------
__global__ void pack_labels_kernel(const int* __restrict__ labels,
                                   unsigned* __restrict__ packed, int N) {
  int i = blockIdx.x * blockDim.x + threadIdx.x;
  if (i < N) {
    unsigned m = 0u;
#pragma unroll
    for (int l = 0; l < L; ++l)
      m |= (labels[i * L + l] != 0) ? (1u << l) : 0u;
    packed[i] = m;
  }
}

// exp(d) with d possibly NaN (from -inf - -inf at startup) -> 0
__device__ __forceinline__ float safe_exp(float d) {
  return (d == d) ? __expf(d) : 0.f;
}

// ---------------------------------------------------------------------------
// Kernel 1: WMMA-tiled E E^T with fused online-softmax row reductions.
// One wave owns a 16-row tile (rows I..I+15), sweeps all column tiles J.
// A-fragment (16x4 f32): lane l, vgpr v -> A[l%16][k + 2*(l/16) + v]
// B-fragment (4x16 f32): lane l, vgpr v -> B[2*(l/16)+v][l%16] = E[J+l%16][..]
//   -> identical per-lane float2 load pattern for both operands (symmetry).
// C/D (16x16 f32): vgpr r, lane l -> row (r + 8*(l/16)), col (l%16).
// ---------------------------------------------------------------------------
__global__ __launch_bounds__(128)
void simloss_kernel(const float* __restrict__ E,
                    const unsigned* __restrict__ packed,
                    float* __restrict__ posExp,
                    float* __restrict__ negExp,
                    unsigned* __restrict__ hasPos,
                    int N) {
  const int lane = threadIdx.x & 31;
  const int wave = threadIdx.x >> 5;
  const int I = (blockIdx.x * 4 + wave) * 16;   // uniform per wave
  if (I >= N) return;

  const int half = lane >> 4;   // 0 | 1
  const int l16  = lane & 15;

  // Cache all A fragments for this wave's 16 rows in registers (64 x v2f).
  v2f aF[KSTEPS];
  const float* arow = E + (size_t)(I + l16) * D + 2 * half;
#pragma unroll
  for (int kk = 0; kk < KSTEPS; ++kk)
    aF[kk] = *reinterpret_cast<const v2f*>(arow + kk * 4);

  // Online-softmax state per (r, half); per-lane partials over lane's columns.
  float m[8], sn[8], sp[8];
  unsigned maskI[8];
  unsigned hpbits = 0u;                 // bit r = row (r,half) saw a positive
#pragma unroll
  for (int r = 0; r < 8; ++r) {
    m[r] = -__builtin_inff();
    sn[r] = 0.f; sp[r] = 0.f;
    maskI[r] = packed[I + r + 8 * half];
  }

  const int ntiles = N >> 4;
  const float* browBase0 = E + (size_t)l16 * D + 2 * half;  // tile 0 B rows

  // Prime the 8-deep B pipeline with tile 0, k-steps 0..7.
  v2f bbuf[PIPE];
  const float* brow = browBase0;
#pragma unroll
  for (int p = 0; p < PIPE; ++p)
    bbuf[p] = *reinterpret_cast<const v2f*>(brow + p * 4);

  for (int jt = 0; jt < ntiles; ++jt) {
    const int J = jt * 16;
    // Next tile's B rows are the next 16 rows of E (wrap to tile 0 at end;
    // the wrapped prefetch is discarded, it just keeps addresses in range).
    const float* brow_next = (jt + 1 < ntiles) ? (brow + 16 * D) : browBase0;

    v8f acc = {0.f, 0.f, 0.f, 0.f, 0.f, 0.f, 0.f, 0.f};
#pragma unroll
    for (int kk = 0; kk < KSTEPS; ++kk) {
      const v2f bcur = bbuf[kk & (PIPE - 1)];
      // Refill this slot PIPE steps ahead (crossing into the next tile).
      const int nk = kk + PIPE;
      const float* src = (nk < KSTEPS) ? (brow + nk * 4)
                                       : (brow_next + (nk - KSTEPS) * 4);
      bbuf[kk & (PIPE - 1)] = *reinterpret_cast<const v2f*>(src);
      acc = __builtin_amdgcn_wmma_f32_16x16x4_f32(
          /*neg_a=*/false, aF[kk], /*neg_b=*/false, bcur,
          /*c_mod=*/(short)0, acc, /*reuse_a=*/false, /*reuse_b=*/false);
    }

    const int gj = J + l16;                       // this lane's column
    const unsigned maskJ = packed[gj];
#pragma unroll
    for (int r = 0; r < 8; ++r) {
      const int gi = I + r + 8 * half;            // this vgpr's row
      const bool offd = (gi != gj);
      const float s = acc[r] * TEMP_INV;
      const float v = offd ? s : -__builtin_inff();
      const float newm = fmaxf(m[r], v);
      const float scale = safe_exp(m[r] - newm);  // 0 at -inf startup
      const float e = offd ? __expf(s - newm) : 0.f;
      const bool pos = offd && ((maskI[r] & maskJ) != 0u);
      sn[r] = sn[r] * scale + e;
      sp[r] = sp[r] * scale + (pos ? e : 0.f);
      hpbits |= pos ? (1u << r) : 0u;
      m[r] = newm;
    }
    brow = brow_next;
  }

  // Merge the 16 lanes of each half (xor offsets <= 8 stay within the half).
#pragma unroll
  for (int off = 1; off < 16; off <<= 1)
    hpbits |= __shfl_xor(hpbits, off, 32);
#pragma unroll
  for (int r = 0; r < 8; ++r) {
#pragma unroll
    for (int off = 1; off < 16; off <<= 1) {
      const float om  = __shfl_xor(m[r], off, 32);
      const float osn = __shfl_xor(sn[r], off, 32);
      const float osp = __shfl_xor(sp[r], off, 32);
      const float M  = fmaxf(m[r], om);
      const float s1 = safe_exp(m[r] - M);
      const float s2 = safe_exp(om - M);
      sn[r] = sn[r] * s1 + osn * s2;
      sp[r] = sp[r] * s1 + osp * s2;
      m[r] = M;
    }
    if (l16 == 0) {
      const int gi = I + r + 8 * half;
      posExp[gi] = sp[r];
      negExp[gi] = sn[r];
      hasPos[gi] = (hpbits >> r) & 1u;
    }
  }
}

// ---------------------------------------------------------------------------
// Kernel 2: final scalar reduction.
// pos/neg are both scaled by exp(-rowmax) identically, so the ratio matches
// the reference's  -log(pos_exp / (neg_exp + 1e-8)).
// ---------------------------------------------------------------------------
__global__ void finalize_kernel(const float* __restrict__ posExp,
                                const float* __restrict__ negExp,
                                const unsigned* __restrict__ hasPos,
                                float* __restrict__ out, int N) {
  __shared__ float ssum[256];
  __shared__ float scnt[256];
  float acc = 0.f, cnt = 0.f;
  for (int i = threadIdx.x; i < N; i += blockDim.x) {
    if (hasPos[i]) {
      acc += logf(negExp[i] + 1e-8f) - logf(posExp[i]);
      cnt += 1.f;
    }
  }
  ssum[threadIdx.x] = acc;
  scnt[threadIdx.x] = cnt;
  __syncthreads();
  for (int s = blockDim.x >> 1; s > 0; s >>= 1) {
    if ((int)threadIdx.x < s) {
      ssum[threadIdx.x] += ssum[threadIdx.x + s];
      scnt[threadIdx.x] += scnt[threadIdx.x + s];
    }
    __syncthreads();
  }
  if (threadIdx.x == 0) out[0] = ssum[0] / fmaxf(scnt[0], 1.f);
}

// ---------------------------------------------------------------------------
extern "C" void kernel_launch(void* const* d_in, const int* in_sizes, int n_in,
                              void* d_out, int out_size, void* d_ws, size_t ws_size,
                              hipStream_t stream) {
  const float* E      = (const float*)d_in[0];
  const int*   labels = (const int*)d_in[1];
  const int N = in_sizes[0] / D;        // 8192

  float* out = (float*)d_out;

  // Workspace carve-up: posExp | negExp | packed | hasPos   (4*N*4 bytes)
  float*    posE   = (float*)d_ws;
  float*    negE   = posE + N;
  unsigned* packed = (unsigned*)(negE + N);
  unsigned* hasP   = packed + N;

  pack_labels_kernel<<<(N + 255) / 256, 256, 0, stream>>>(labels, packed, N);

  const int tileBlocks = ((N / 16) + 3) / 4;   // 4 waves/block, 1 tile/wave
  simloss_kernel<<<tileBlocks, 128, 0, stream>>>(
      E, packed, posE, negE, hasP, N);

  finalize_kernel<<<1, 256, 0, stream>>>(posE, negE, hasP, out, N);
}